// RGBDemosaic_45956150067559
// MI455X (gfx1250) — compile-verified
//
#include <hip/hip_runtime.h>
#include <stdint.h>

// Problem constants (from reference): B=64, T=14, planes=4, H=W=96 -> out [64,3,384,384]
#define BATCH   64
#define TFRM    14
#define HH      96
#define WW      96
#define DEM     192      // demosaiced resolution (2H)
#define OUTSZ   384      // final resolution
#define TY      16       // output rows per block
#define NTILES  (OUTSZ / TY)   // 24 row tiles
#define THREADS 256      // 8 wave32 waves
#define WIN     6        // staged plane rows per tile
#define DROWS   10       // demosaic rows per tile

typedef float v4f __attribute__((ext_vector_type(4)));

// Fused demosaic + 2x bilinear resize.
//   Phase 1: gfx1250 async global->LDS DMA (ASYNCcnt path) stages the Bayer window.
//   Phase 2: renormalized-border demosaic of the 10 needed rows into LDS.
//   Phase 3: 2x bilinear blend, 4 pixels/thread, non-temporal b128 stores.
__global__ __launch_bounds__(THREADS)
void demosaic_resize_kernel(const float* __restrict__ in, float* __restrict__ out)
{
    __shared__ float splane[4][WIN][WW];     //  9216 B : staged Bayer plane rows
    __shared__ float sdem[3][DROWS][DEM];    // 23040 B : demosaiced RGB rows

    const int t   = blockIdx.x;              // row tile [0,24)
    const int b   = blockIdx.y;              // batch    [0,64)
    const int tid = threadIdx.x;

    // ---------------- Phase 1: async-stage plane rows [r0, r0+6) ----------------
    const int r0 = min(max(4 * t - 1, 0), HH - WIN);
    const uint64_t frame = (uint64_t)(uintptr_t)in
                         + (uint64_t)b * (uint64_t)(TFRM * 4 * HH * WW) * 4ull; // frame 0 of batch b
    const uint32_t lds0  = (uint32_t)(uintptr_t)(&splane[0][0][0]);             // low 32 bits = LDS offset

    // 4 planes * 6 rows * 96 floats = 576 chunks of 16 B, plane-major then row.
    for (int c = tid; c < 4 * WIN * (WW / 4); c += THREADS) {
        int row = c / (WW / 4);                 // 0..23
        int p   = row / WIN;                    // plane
        int wr  = row % WIN;                    // row within window
        int xo  = (c % (WW / 4)) * 16;          // byte offset within row
        uint64_t ga = frame + (uint64_t)p * (HH * WW * 4)
                            + (uint64_t)(r0 + wr) * (WW * 4) + (uint64_t)xo;
        uint32_t la = lds0 + (uint32_t)c * 16u;
        asm volatile("global_load_async_to_lds_b128 %0, %1, off"
                     :: "v"(la), "v"(ga) : "memory");
    }
    asm volatile("s_wait_asynccnt 0x0" ::: "memory");  // per-wave: own DMAs landed in LDS
    __syncthreads();                                    // cross-wave visibility

    // ---------------- Phase 2: demosaic rows j0..j0+9 into LDS ----------------
    const int j0 = 8 * t - 1;

    auto ld = [&](int p, int r, int c_) -> float { return splane[p][r - r0][c_]; };
    #define ACC(nm, dn, p, r, c_, w)                                             \
        do { int _r = (r), _c = (c_);                                            \
             if (_r >= 0 && _r < HH && _c >= 0 && _c < WW) {                     \
                 nm += (w) * ld(p, _r, _c); dn += (w);                           \
             } } while (0)

    for (int idx = tid; idx < DROWS * DEM; idx += THREADS) {
        int dj = idx / DEM;
        int i  = idx % DEM;
        int j  = min(max(j0 + dj, 0), DEM - 1);
        int hy = j >> 1, hx = i >> 1;
        int py = j & 1,  px = i & 1;

        float rn = 0.f, rd = 0.f, gn = 0.f, gd = 0.f, bn = 0.f, bd = 0.f;

        // R (plane 0 lives at even/even full-res sites)
        if (py == 0 && px == 0)      { rn = ld(0, hy, hx); rd = 1.f; }
        else if (py == 0)            { ACC(rn, rd, 0, hy, hx, 0.5f);  ACC(rn, rd, 0, hy, hx + 1, 0.5f); }
        else if (px == 0)            { ACC(rn, rd, 0, hy, hx, 0.5f);  ACC(rn, rd, 0, hy + 1, hx, 0.5f); }
        else { ACC(rn, rd, 0, hy, hx, 0.25f);     ACC(rn, rd, 0, hy, hx + 1, 0.25f);
               ACC(rn, rd, 0, hy + 1, hx, 0.25f); ACC(rn, rd, 0, hy + 1, hx + 1, 0.25f); }

        // B (plane 3 lives at odd/odd sites)
        if (py == 1 && px == 1)      { bn = ld(3, hy, hx); bd = 1.f; }
        else if (py == 1)            { ACC(bn, bd, 3, hy, hx - 1, 0.5f); ACC(bn, bd, 3, hy, hx, 0.5f); }
        else if (px == 1)            { ACC(bn, bd, 3, hy - 1, hx, 0.5f); ACC(bn, bd, 3, hy, hx, 0.5f); }
        else { ACC(bn, bd, 3, hy - 1, hx - 1, 0.25f); ACC(bn, bd, 3, hy - 1, hx, 0.25f);
               ACC(bn, bd, 3, hy, hx - 1, 0.25f);     ACC(bn, bd, 3, hy, hx, 0.25f); }

        // G (plane 1 at even/odd, plane 2 at odd/even)
        if (py == 0 && px == 1)      { gn = ld(1, hy, hx); gd = 1.f; }
        else if (py == 1 && px == 0) { gn = ld(2, hy, hx); gd = 1.f; }
        else if (py == 0) {          // R site: cross of 4 G neighbors
            ACC(gn, gd, 1, hy, hx - 1, 0.5f); ACC(gn, gd, 1, hy, hx, 0.5f);
            ACC(gn, gd, 2, hy - 1, hx, 0.5f); ACC(gn, gd, 2, hy, hx, 0.5f);
        } else {                     // B site
            ACC(gn, gd, 2, hy, hx, 0.5f);     ACC(gn, gd, 2, hy, hx + 1, 0.5f);
            ACC(gn, gd, 1, hy, hx, 0.5f);     ACC(gn, gd, 1, hy + 1, hx, 0.5f);
        }

        sdem[0][dj][i] = rn / rd;
        sdem[1][dj][i] = gn / gd;
        sdem[2][dj][i] = bn / bd;
    }
    __syncthreads();

    // ------- Phase 3: 2x bilinear, 4 consecutive pixels/thread, NT b128 stores -------
    float* outb = out + (uint64_t)b * 3ull * (OUTSZ * OUTSZ);
    const int ybase = TY * t;

    // 16 rows * 96 four-pixel groups = 1536 groups = exactly 6 per thread.
    for (int idx = tid; idx < TY * (OUTSZ / 4); idx += THREADS) {
        int yl = idx / (OUTSZ / 4);
        int xc = idx % (OUTSZ / 4);
        int y  = ybase + yl;

        // vertical taps: coord = y/2 - 0.25 (half-pixel mapping), edge-renormalized
        int   k = y >> 1;
        int   rlo, rhi; float wlo, whi;
        if ((y & 1) == 0) { rlo = k - 1; rhi = k;     wlo = 0.25f; whi = 0.75f;
                            if (rlo < 0)       { rlo = 0;       wlo = 0.f; whi = 1.f; } }
        else              { rlo = k;     rhi = k + 1; wlo = 0.75f; whi = 0.25f;
                            if (rhi > DEM - 1) { rhi = DEM - 1; whi = 0.f; wlo = 1.f; } }
        const int dlo = rlo - j0, dhi = rhi - j0;    // in [0, DROWS)

        // 4 output pixels x0..x0+3 (x0 = 4*xc) touch dem columns {cA-1, cA, cB, cB+1}
        const int cA = 2 * xc, cB = cA + 1;
        const int   cm1 = (cA > 0)       ? cA - 1 : 0;
        const int   cp2 = (cB < DEM - 1) ? cB + 1 : DEM - 1;
        const float w0a = (cA > 0)       ? 0.25f : 0.f;   // pixel0: even-x taps (0.25, 0.75)
        const float w0b = (cA > 0)       ? 0.75f : 1.f;
        const float w3e = (cB < DEM - 1) ? 0.25f : 0.f;   // pixel3: odd-x taps (0.75, 0.25)
        const float w3b = (cB < DEM - 1) ? 0.75f : 1.f;

        #pragma unroll
        for (int ch = 0; ch < 3; ++ch) {
            const float aL = sdem[ch][dlo][cm1], b0L = sdem[ch][dlo][cA],
                        b1L = sdem[ch][dlo][cB], eL  = sdem[ch][dlo][cp2];
            const float aH = sdem[ch][dhi][cm1], b0H = sdem[ch][dhi][cA],
                        b1H = sdem[ch][dhi][cB], eH  = sdem[ch][dhi][cp2];

            // horizontal interp per dem row (shared across the 4 outputs)
            const float lo0 = w0a * aL + w0b * b0L;
            const float lo1 = 0.75f * b0L + 0.25f * b1L;
            const float lo2 = 0.25f * b0L + 0.75f * b1L;
            const float lo3 = w3b * b1L + w3e * eL;
            const float hi0 = w0a * aH + w0b * b0H;
            const float hi1 = 0.75f * b0H + 0.25f * b1H;
            const float hi2 = 0.25f * b0H + 0.75f * b1H;
            const float hi3 = w3b * b1H + w3e * eH;

            v4f v;
            v.x = wlo * lo0 + whi * hi0;
            v.y = wlo * lo1 + whi * hi1;
            v.z = wlo * lo2 + whi * hi2;
            v.w = wlo * lo3 + whi * hi3;

            v4f* dst = (v4f*)(outb + (uint64_t)ch * (OUTSZ * OUTSZ)
                                   + (uint64_t)y * OUTSZ + (uint64_t)(4 * xc));
            __builtin_nontemporal_store(v, dst);   // write-once data: th:NT, keep L2 for input
        }
    }
}

extern "C" void kernel_launch(void* const* d_in, const int* in_sizes, int n_in,
                              void* d_out, int out_size, void* d_ws, size_t ws_size,
                              hipStream_t stream)
{
    const float* in  = (const float*)d_in[0];   // [64,14,4,96,96] f32; only frame 0 read
    float*       out = (float*)d_out;           // [64,3,384,384] f32

    dim3 grid(NTILES, BATCH, 1);                // 24 x 64 = 1536 blocks
    hipLaunchKernelGGL(demosaic_resize_kernel, grid, dim3(THREADS, 1, 1), 0, stream, in, out);
}